// S4DKernel_90426241449996
// MI455X (gfx1250) — compile-verified
//
#include <hip/hip_runtime.h>
#include <hip/hip_bf16.h>
#include <math.h>

// ---------------------------------------------------------------------------
// S4D kernel materialization for gfx1250 (MI455X).
//
//   K[l] = sum_n Re( Cc_n * exp(dtA_n * l) ),   l in [0, 65536), n in [0,1024)
//
// Factor l = r*256 + c:
//   V[n,l] = P[n,r] * Q[n,c],  P = exp(dtA*256r), Q = exp(dtA*c)
//   W[n,r] = Cc_n * P[n,r]
//   K[r*256+c] = sum_n ( Wr[n,r]*Qr[n,c] - Wi[n,r]*Qi[n,c] )
// => real GEMM:  D(256x256) = A(256x2048) * B(2048x256)
//    A[r, n     ] =  Wr[n,r]     B[n,      c] = Qr[n,c]
//    A[r, n+1024] = -Wi[n,r]     B[n+1024, c] = Qi[n,c]
// GEMM runs on V_WMMA_F32_16X16X4_F32 (full fp32 — accumulation has heavy
// sinusoidal cancellation; fp32 is free here, the problem is tiny).
// ---------------------------------------------------------------------------

typedef __attribute__((ext_vector_type(2))) float v2f;
typedef __attribute__((ext_vector_type(8))) float v8f;

#define NMODES 1024
#define TLO    256     // low-l tile width  (c)
#define RHI    256     // high-l tile count (r)
#define KDIM   (2 * NMODES)

// ---------------------------------------------------------------------------
// Pass 1: build the GEMM operands (K-major so WMMA fragment loads coalesce).
//   Abuf[k][m] : k = n (Wr row) or n+1024 (-Wi row), m = r    (2048 x 256)
//   Bbuf[k][c] : k = n (Qr row) or n+1024 ( Qi row), c = l_lo (2048 x 256)
// One block per mode n, one thread per (r == c == t).
// ---------------------------------------------------------------------------
__global__ __launch_bounds__(256) void s4d_build_tables(
    const float* __restrict__ C,        // (N,2)
    const float* __restrict__ log_step, // (1,)
    const float* __restrict__ A_re,     // (N,)
    const float* __restrict__ A_im,     // (N,)
    float* __restrict__ Abuf,           // (2048, 256)
    float* __restrict__ Bbuf)           // (2048, 256)
{
    const int n = blockIdx.x;     // mode
    const int t = threadIdx.x;    // r for A-rows, c for B-rows

    const float dt = expf(log_step[0]);
    const float arA = fminf(A_re[n], -1e-4f);  // clamped real part of A
    const float aiA = A_im[n];
    const float da = dt * arA;                 // Re(dtA)
    const float db = dt * aiA;                 // Im(dtA)

    // Discretized C:  Cc = (C0 + i C1) * (exp(dtA) - 1) / A
    const float e0 = expf(da);
    float s0, c0;
    sincosf(db, &s0, &c0);
    const float numr = e0 * c0 - 1.0f;
    const float numi = e0 * s0;
    const float inv_a2 = 1.0f / (arA * arA + aiA * aiA);
    const float fr = (numr * arA + numi * aiA) * inv_a2;   // (exp(dtA)-1)/A
    const float fi = (numi * arA - numr * aiA) * inv_a2;
    const float C0 = C[2 * n + 0];
    const float C1 = C[2 * n + 1];
    const float Ccr = C0 * fr - C1 * fi;
    const float Cci = C0 * fi + C1 * fr;

    // B rows: Q[n, c] = exp(dtA * c), c = t
    {
        const float l = (float)t;
        const float e = expf(da * l);
        float s, c;
        sincosf(db * l, &s, &c);
        Bbuf[n * TLO + t]            = e * c;   // Qr
        Bbuf[(n + NMODES) * TLO + t] = e * s;   // Qi
    }

    // A rows: W[n, r] = Cc_n * exp(dtA * 256r), r = t
    {
        const float l = (float)(t * TLO);
        const float e = expf(da * l);
        float s, c;
        sincosf(db * l, &s, &c);
        const float Pr = e * c;
        const float Pi = e * s;
        const float Wr = Ccr * Pr - Cci * Pi;
        const float Wi = Ccr * Pi + Cci * Pr;
        Abuf[n * RHI + t]            =  Wr;
        Abuf[(n + NMODES) * RHI + t] = -Wi;     // sign fold for the Im product
    }
}

// ---------------------------------------------------------------------------
// Pass 2: D(256x256) = A(256x2048) x B(2048x256) on V_WMMA_F32_16X16X4_F32.
// One wave32 per 16x16 output tile; 256 waves total (4 waves / block).
// A fragment (16x4 f32, 2 VGPR): lanes 0-15 hold K=k,k+1; lanes 16-31 K=k+2,k+3.
// C/D (16x16 f32, 8 VGPR): VGPR v = rows v (lanes 0-15) and v+8 (lanes 16-31).
// ---------------------------------------------------------------------------
__global__ __launch_bounds__(128) void s4d_wmma_gemm(
    const float* __restrict__ Abuf,   // (KDIM, 256)  [k][m]
    const float* __restrict__ Bbuf,   // (KDIM, 256)  [k][c]
    float* __restrict__ out)          // (256, 256) == K[0:65536]
{
    const int lane = threadIdx.x & 31;
    const int wave = threadIdx.x >> 5;
    const int tile = blockIdx.x * 4 + wave;     // 0..255
    const int tm   = (tile >> 4) * 16;          // output row base  (r)
    const int tn   = (tile & 15) * 16;          // output col base  (c)
    const int half = lane >> 4;                 // 0: K lo pair, 1: K hi pair
    const int l16  = lane & 15;
    const int m    = tm + l16;                  // A fragment row
    const int nn   = tn + l16;                  // B fragment col

    v8f acc = {};

    for (int k0 = 0; k0 < KDIM; k0 += 64) {
        // speculative prefetch of the next K-chunk (-> global_prefetch_b8)
        __builtin_prefetch(&Abuf[(k0 + 64) * RHI + m], 0, 1);
        __builtin_prefetch(&Bbuf[(k0 + 64) * TLO + nn], 0, 1);
#pragma unroll
        for (int kk = 0; kk < 64; kk += 4) {
            const int kb = k0 + kk + half * 2;
            v2f a, b;
            a.x = Abuf[(kb + 0) * RHI + m];
            a.y = Abuf[(kb + 1) * RHI + m];
            b.x = Bbuf[(kb + 0) * TLO + nn];
            b.y = Bbuf[(kb + 1) * TLO + nn];
            // (neg_a, A, neg_b, B, c_mod, C, reuse_a, reuse_b)
            acc = __builtin_amdgcn_wmma_f32_16x16x4_f32(
                false, a, false, b, (short)0, acc, false, false);
        }
    }

#pragma unroll
    for (int v = 0; v < 8; ++v) {
        const int row = tm + v + half * 8;
        out[row * TLO + nn] = acc[v];
    }
}

// ---------------------------------------------------------------------------
extern "C" void kernel_launch(void* const* d_in, const int* in_sizes, int n_in,
                              void* d_out, int out_size, void* d_ws, size_t ws_size,
                              hipStream_t stream) {
    const float* C        = (const float*)d_in[0];  // (1024, 2)
    const float* log_step = (const float*)d_in[1];  // (1,)
    const float* A_re     = (const float*)d_in[2];  // (1024,)
    const float* A_im     = (const float*)d_in[3];  // (1024,)
    // d_in[4] is L (== 65536), compile-time constant here.

    float* Abuf = (float*)d_ws;                     // 2048*256 f32 = 2 MB
    float* Bbuf = Abuf + KDIM * RHI;                // 2048*256 f32 = 2 MB
    float* out  = (float*)d_out;                    // 65536 f32

    s4d_build_tables<<<NMODES, 256, 0, stream>>>(C, log_step, A_re, A_im, Abuf, Bbuf);
    s4d_wmma_gemm<<<64, 128, 0, stream>>>(Abuf, Bbuf, out);
}